// KGNNModel_10548439679326
// MI455X (gfx1250) — compile-verified
//
#include <hip/hip_runtime.h>
#include <math.h>

typedef __attribute__((ext_vector_type(2))) float v2f;
typedef __attribute__((ext_vector_type(8))) float v8f;

#define TPB 256

__device__ __forceinline__ float sigm(float x) { return 1.0f / (1.0f + expf(-x)); }

// order-preserving float<->uint for atomicMax-based segment max
__device__ __forceinline__ unsigned f2ord(float f) {
  unsigned u = __float_as_uint(f);
  return (u & 0x80000000u) ? ~u : (u | 0x80000000u);
}
__device__ __forceinline__ float ord2f(unsigned u) {
  return (u & 0x80000000u) ? __uint_as_float(u & 0x7FFFFFFFu) : __uint_as_float(~u);
}

// -------------------------------------------------------------------------
// Utility kernels
// -------------------------------------------------------------------------
__global__ void k_zero(float* __restrict__ p, size_t n) {
  size_t i = (size_t)blockIdx.x * TPB + threadIdx.x;
  if (i < n) p[i] = 0.0f;
}
__global__ void k_zero_u(unsigned* __restrict__ p, int n) {
  int i = blockIdx.x * TPB + threadIdx.x;
  if (i < n) p[i] = 0u;
}
__global__ void k_relu(float* __restrict__ p, size_t n) {
  size_t i = (size_t)blockIdx.x * TPB + threadIdx.x;
  if (i < n) p[i] = fmaxf(p[i], 0.0f);
}

// edge type = argmax(edge_attr[e, 0:4]) (first-max wins, as jnp.argmax)
__global__ void k_etype(const float* __restrict__ ea, int* __restrict__ et, int E) {
  int e = blockIdx.x * TPB + threadIdx.x;
  if (e >= E) return;
  const float* row = ea + (size_t)e * 10;
  int best = 0; float bv = row[0];
#pragma unroll
  for (int i = 1; i < 4; ++i) { float v = row[i]; if (v > bv) { bv = v; best = i; } }
  et[e] = best;
}

// -------------------------------------------------------------------------
// WMMA fp32 GEMM: tr[t] = h @ W_t   (h:[Nn,64], W_t:[64,64] row-major c x d)
// one wave per 16x16 tile; K stepped by 4 using V_WMMA_F32_16X16X4_F32
// A 16x4 layout: lanes0-15 row=M0+l16 hold K=k0,k0+1 ; lanes16-31 hold k0+2,k0+3
// B 4x16 layout: v[0]: lanes0-15 B[k0][col], lanes16-31 B[k0+2][col]; v[1]: k0+1/k0+3
// -------------------------------------------------------------------------
__global__ void k_transform(const float* __restrict__ h, const float* __restrict__ Wl,
                            float* __restrict__ tr, int Nn) {
  const int lane = threadIdx.x;
  const int half = lane >> 4, l16 = lane & 15;
  const int M0 = blockIdx.x * 16, N0 = blockIdx.y * 16, t = blockIdx.z;
  const float* W = Wl + (size_t)t * 64 * 64;
  int rowA = M0 + l16; if (rowA >= Nn) rowA = Nn - 1;
  const float* arow = h + (size_t)rowA * 64;
  v8f acc = {};
#pragma unroll
  for (int k0 = 0; k0 < 64; k0 += 4) {
    int ka = k0 + 2 * half;
    v2f a = *(const v2f*)(arow + ka);
    v2f b;
    b[0] = W[(size_t)ka * 64 + N0 + l16];
    b[1] = W[(size_t)(ka + 1) * 64 + N0 + l16];
    acc = __builtin_amdgcn_wmma_f32_16x16x4_f32(false, a, false, b, (short)0, acc,
                                                false, false);
  }
  float* out = tr + ((size_t)t * Nn + 0) * 64;
#pragma unroll
  for (int r = 0; r < 8; ++r) {
    int row = M0 + r + 8 * half;
    if (row < Nn) out[(size_t)row * 64 + N0 + l16] = acc[r];
  }
}

// -------------------------------------------------------------------------
// Edge scatter: aggr[dst] += tr[etype][src] * coeff(src)
// 16 threads per edge, float4 per thread, native fp32 atomics
// -------------------------------------------------------------------------
__global__ void k_scatter(const float* __restrict__ tr, const int* __restrict__ src,
                          const int* __restrict__ dst, const int* __restrict__ etype,
                          const float* __restrict__ ntRow, const int* __restrict__ ntype,
                          float* __restrict__ aggr, int E, int Nn) {
  int gid = blockIdx.x * TPB + threadIdx.x;
  int e = gid >> 4;
  int g = gid & 15;
  if (e >= E) return;
  int s = src[e], d = dst[e];
  int t = etype ? etype[e] : 0;
  float cf = ntRow ? ntRow[ntype[s]] : 1.0f;
  const float4 v = *reinterpret_cast<const float4*>(tr + ((size_t)t * Nn + s) * 64 + g * 4);
  float* out = aggr + (size_t)d * 64 + g * 4;
  unsafeAtomicAdd(out + 0, v.x * cf);
  unsafeAtomicAdd(out + 1, v.y * cf);
  unsafeAtomicAdd(out + 2, v.z * cf);
  unsafeAtomicAdd(out + 3, v.w * cf);
}

// -------------------------------------------------------------------------
// Fused GRU: h' = GRU(aggr, h). 6 WMMA accumulators: gi/gh x {r,z,n}.
// wih/whh: [192,64] row-major; gi[n,j] = sum_c aggr[n,c]*wih[j,c]
// one wave per (16 rows x 16 gate-cols), grid.y in 0..3 covers cols 0..63
// -------------------------------------------------------------------------
__global__ void k_gru(const float* __restrict__ aggr, const float* __restrict__ h,
                      const float* __restrict__ wih, const float* __restrict__ whh,
                      const float* __restrict__ bih, const float* __restrict__ bhh,
                      float* __restrict__ hout, int Nn) {
  const int lane = threadIdx.x;
  const int half = lane >> 4, l16 = lane & 15;
  const int M0 = blockIdx.x * 16, N0 = blockIdx.y * 16;
  int rowA = M0 + l16; if (rowA >= Nn) rowA = Nn - 1;
  const float* arow = aggr + (size_t)rowA * 64;
  const float* hrow = h + (size_t)rowA * 64;
  const int jr = N0 + l16, jz = jr + 64, jn = jr + 128;
  const float* wi_r = wih + (size_t)jr * 64;
  const float* wi_z = wih + (size_t)jz * 64;
  const float* wi_n = wih + (size_t)jn * 64;
  const float* wh_r = whh + (size_t)jr * 64;
  const float* wh_z = whh + (size_t)jz * 64;
  const float* wh_n = whh + (size_t)jn * 64;
  v8f air = {}, aiz = {}, ain = {}, ahr = {}, ahz = {}, ahn = {};
#pragma unroll
  for (int k0 = 0; k0 < 64; k0 += 4) {
    int ka = k0 + 2 * half;
    v2f aA = *(const v2f*)(arow + ka);
    v2f aH = *(const v2f*)(hrow + ka);
    air = __builtin_amdgcn_wmma_f32_16x16x4_f32(false, aA, false, *(const v2f*)(wi_r + ka),
                                                (short)0, air, false, false);
    aiz = __builtin_amdgcn_wmma_f32_16x16x4_f32(false, aA, false, *(const v2f*)(wi_z + ka),
                                                (short)0, aiz, false, false);
    ain = __builtin_amdgcn_wmma_f32_16x16x4_f32(false, aA, false, *(const v2f*)(wi_n + ka),
                                                (short)0, ain, false, false);
    ahr = __builtin_amdgcn_wmma_f32_16x16x4_f32(false, aH, false, *(const v2f*)(wh_r + ka),
                                                (short)0, ahr, false, false);
    ahz = __builtin_amdgcn_wmma_f32_16x16x4_f32(false, aH, false, *(const v2f*)(wh_z + ka),
                                                (short)0, ahz, false, false);
    ahn = __builtin_amdgcn_wmma_f32_16x16x4_f32(false, aH, false, *(const v2f*)(wh_n + ka),
                                                (short)0, ahn, false, false);
  }
  const int col = N0 + l16;
  const float bir = bih[col], biz = bih[col + 64], bin = bih[col + 128];
  const float bhr = bhh[col], bhz = bhh[col + 64], bhn = bhh[col + 128];
#pragma unroll
  for (int r = 0; r < 8; ++r) {
    int row = M0 + r + 8 * half;
    if (row >= Nn) continue;
    float ir = air[r] + bir, hr = ahr[r] + bhr;
    float iz = aiz[r] + biz, hz = ahz[r] + bhz;
    float in_ = ain[r] + bin, hn = ahn[r] + bhn;
    float rg = sigm(ir + hr);
    float zg = sigm(iz + hz);
    float ng = tanhf(in_ + rg * hn);
    float hold = h[(size_t)row * 64 + col];
    hout[(size_t)row * 64 + col] = (1.0f - zg) * ng + zg * hold;
  }
}

// -------------------------------------------------------------------------
// BatchNorm over rows (axis 0): per-column mean / rsqrt(var+eps)
// -------------------------------------------------------------------------
__global__ void k_colstats(const float* __restrict__ x, int rows, int cols,
                           float* __restrict__ mu, float* __restrict__ rs) {
  __shared__ float s1[TPB], s2[TPB];
  int c = blockIdx.x, t = threadIdx.x;
  float a = 0.0f, b = 0.0f;
  for (int r = t; r < rows; r += TPB) {
    float v = x[(size_t)r * cols + c];
    a += v; b += v * v;
  }
  s1[t] = a; s2[t] = b;
  __syncthreads();
  for (int s = TPB / 2; s > 0; s >>= 1) {
    if (t < s) { s1[t] += s1[t + s]; s2[t] += s2[t + s]; }
    __syncthreads();
  }
  if (t == 0) {
    float m = s1[0] / (float)rows;
    float v = s2[0] / (float)rows - m * m;
    mu[c] = m;
    rs[c] = rsqrtf(fmaxf(v, 0.0f) + 1e-5f);
  }
}

__global__ void k_bn_apply(float* __restrict__ x, const float* __restrict__ mu,
                           const float* __restrict__ rs, const float* __restrict__ g,
                           const float* __restrict__ b, int relu, size_t n, int cols) {
  size_t i = (size_t)blockIdx.x * TPB + threadIdx.x;
  if (i >= n) return;
  int c = (int)(i % (size_t)cols);
  float v = (x[i] - mu[c]) * rs[c] * g[c] + b[c];
  if (relu) v = fmaxf(v, 0.0f);
  x[i] = v;
}

// -------------------------------------------------------------------------
// Set2Set building blocks
// -------------------------------------------------------------------------
// one block per graph; 256 threads compute 256 gate values, then 64 update h,c
__global__ void k_lstm(const float* __restrict__ inp, int Kin,
                       float* __restrict__ h, float* __restrict__ c,
                       const float* __restrict__ wih, const float* __restrict__ whh,
                       const float* __restrict__ bias) {
  __shared__ float gates[256];
  int b = blockIdx.x, j = threadIdx.x;
  float acc = bias[j];
  const float* irow = inp + (size_t)b * Kin;
  const float* wr = wih + (size_t)j * Kin;
  for (int k = 0; k < Kin; ++k) acc += irow[k] * wr[k];
  const float* hrow = h + (size_t)b * 64;
  const float* wh = whh + (size_t)j * 64;
  for (int k = 0; k < 64; ++k) acc += hrow[k] * wh[k];
  gates[j] = acc;
  __syncthreads();
  if (j < 64) {
    float ig = gates[j], fg = gates[j + 64], gg = gates[j + 128], og = gates[j + 192];
    float cv = sigm(fg) * c[(size_t)b * 64 + j] + sigm(ig) * tanhf(gg);
    c[(size_t)b * 64 + j] = cv;
    h[(size_t)b * 64 + j] = sigm(og) * tanhf(cv);
  }
}

__global__ void k_att_dot(const float* __restrict__ x, const float* __restrict__ q,
                          const int* __restrict__ batch, float* __restrict__ e, int Nn) {
  int n = blockIdx.x * TPB + threadIdx.x;
  if (n >= Nn) return;
  const float* xr = x + (size_t)n * 64;
  const float* qr = q + (size_t)batch[n] * 64;
  float s = 0.0f;
#pragma unroll 4
  for (int cc = 0; cc < 64; ++cc) s += xr[cc] * qr[cc];
  e[n] = s;
}

__global__ void k_segmax(const float* __restrict__ e, const int* __restrict__ batch,
                         unsigned* __restrict__ emax, int Nn) {
  int n = blockIdx.x * TPB + threadIdx.x;
  if (n >= Nn) return;
  atomicMax(&emax[batch[n]], f2ord(e[n]));
}

__global__ void k_expden(const float* __restrict__ e, const unsigned* __restrict__ emax,
                         const int* __restrict__ batch, float* __restrict__ ex,
                         float* __restrict__ denom, int Nn) {
  int n = blockIdx.x * TPB + threadIdx.x;
  if (n >= Nn) return;
  float v = expf(e[n] - ord2f(emax[batch[n]]));
  ex[n] = v;
  unsafeAtomicAdd(&denom[batch[n]], v);
}

__global__ void k_attw(const float* __restrict__ ex, const float* __restrict__ denom,
                       const int* __restrict__ batch, float* __restrict__ a, int Nn) {
  int n = blockIdx.x * TPB + threadIdx.x;
  if (n >= Nn) return;
  a[n] = ex[n] / denom[batch[n]];
}

__global__ void k_wsum(const float* __restrict__ a, const float* __restrict__ x,
                       const int* __restrict__ batch, float* __restrict__ r, int Nn) {
  size_t id = (size_t)blockIdx.x * TPB + threadIdx.x;
  if (id >= (size_t)Nn * 64) return;
  int n = (int)(id >> 6), cc = (int)(id & 63);
  unsafeAtomicAdd(&r[(size_t)batch[n] * 64 + cc], a[n] * x[(size_t)n * 64 + cc]);
}

__global__ void k_qstar(const float* __restrict__ h1, const float* __restrict__ r,
                        float* __restrict__ qstar, int B) {
  int id = blockIdx.x * TPB + threadIdx.x;
  if (id >= B * 128) return;
  int b = id >> 7, k = id & 127;
  qstar[id] = (k < 64) ? h1[(size_t)b * 64 + k] : r[(size_t)b * 64 + (k - 64)];
}

// -------------------------------------------------------------------------
// avg_pool level1 -> level2
// -------------------------------------------------------------------------
__global__ void k_apool_scatter(const float* __restrict__ h, const int* __restrict__ asrc,
                                const int* __restrict__ adst, float* __restrict__ hsum,
                                float* __restrict__ cnt, int A) {
  size_t id = (size_t)blockIdx.x * TPB + threadIdx.x;
  if (id >= (size_t)A * 64) return;
  int a = (int)(id >> 6), cc = (int)(id & 63);
  int s = asrc[a], d = adst[a];
  unsafeAtomicAdd(&hsum[(size_t)d * 64 + cc], h[(size_t)s * 64 + cc]);
  if (cc == 0) unsafeAtomicAdd(&cnt[d], 1.0f);
}

__global__ void k_apool_div(float* __restrict__ h2, const float* __restrict__ hsum,
                            const float* __restrict__ cnt, int N2) {
  size_t id = (size_t)blockIdx.x * TPB + threadIdx.x;
  if (id >= (size_t)N2 * 64) return;
  h2[id] = hsum[id] / fmaxf(cnt[id >> 6], 1.0f);
}

// -------------------------------------------------------------------------
// FC head helpers
// -------------------------------------------------------------------------
__global__ void k_concat(const float* __restrict__ x1, const float* __restrict__ x2,
                         float* __restrict__ z, int B) {
  int id = blockIdx.x * TPB + threadIdx.x;
  if (id >= B * 256) return;
  int b = id >> 8, k = id & 255;
  z[id] = (k < 128) ? x1[(size_t)b * 128 + k] : x2[(size_t)b * 128 + (k - 128)];
}

__global__ void k_fc(const float* __restrict__ in, const float* __restrict__ wt,
                     const float* __restrict__ bias, float* __restrict__ out,
                     int R, int K, int Cout) {
  int id = blockIdx.x * TPB + threadIdx.x;
  if (id >= R * Cout) return;
  int r = id / Cout, j = id % Cout;
  float s = bias[j];
  const float* ir = in + (size_t)r * K;
  const float* wr = wt + (size_t)j * K;
  for (int k = 0; k < K; ++k) s += ir[k] * wr[k];
  out[id] = s;
}

// -------------------------------------------------------------------------
// Host orchestration
// -------------------------------------------------------------------------
static inline unsigned gdiv(size_t n, unsigned b) { return (unsigned)((n + b - 1) / b); }

extern "C" void kernel_launch(void* const* d_in, const int* in_sizes, int n_in,
                              void* d_out, int out_size, void* d_ws, size_t ws_size,
                              hipStream_t stream) {
  (void)n_in; (void)out_size; (void)ws_size;
  const float* x        = (const float*)d_in[0];
  const float* edgeattr = (const float*)d_in[1];
  const float* conv_w   = (const float*)d_in[2];
  const float* conv_wih = (const float*)d_in[3];
  const float* conv_whh = (const float*)d_in[4];
  const float* conv_bih = (const float*)d_in[5];
  const float* conv_bhh = (const float*)d_in[6];
  const float* conv_nt  = (const float*)d_in[7];
  const float* s2s_w[2][6] = {
    {(const float*)d_in[8], (const float*)d_in[9], (const float*)d_in[10],
     (const float*)d_in[11], (const float*)d_in[12], (const float*)d_in[13]},
    {(const float*)d_in[14], (const float*)d_in[15], (const float*)d_in[16],
     (const float*)d_in[17], (const float*)d_in[18], (const float*)d_in[19]}};
  const float* bn_g     = (const float*)d_in[20];
  const float* bn_b     = (const float*)d_in[21];
  const float* prebn_g  = (const float*)d_in[22];
  const float* prebn_b  = (const float*)d_in[23];
  const float* fc_w0    = (const float*)d_in[24];
  const float* fc_b0    = (const float*)d_in[25];
  const float* fc_w1    = (const float*)d_in[26];
  const float* fc_b1    = (const float*)d_in[27];
  const float* fc_w2    = (const float*)d_in[28];
  const float* fc_b2    = (const float*)d_in[29];
  const float* fcbn_g0  = (const float*)d_in[30];
  const float* fcbn_b0  = (const float*)d_in[31];
  const float* fcbn_g1  = (const float*)d_in[32];
  const float* fcbn_b1  = (const float*)d_in[33];
  const int* edge_index = (const int*)d_in[34];
  const int* node_type  = (const int*)d_in[35];
  const int* batch      = (const int*)d_in[36];
  const int* assign_src = (const int*)d_in[37];
  const int* assign_dst = (const int*)d_in[38];
  const int* edge_index2 = (const int*)d_in[39];
  const int* batch2     = (const int*)d_in[40];

  const int N  = in_sizes[0] / 64;
  const int E  = in_sizes[34] / 2;
  const int A  = in_sizes[37];
  const int N2 = in_sizes[40];
  const int E2 = in_sizes[39] / 2;
  const int B  = 256;  // num_graphs (reference setup constant; device scalar at d_in[41])

  const int* srcP = edge_index;
  const int* dstP = edge_index + E;
  const int* src2 = edge_index2;
  const int* dst2 = edge_index2 + E2;

  // ---- workspace layout (bump allocator, 256B aligned) ----
  char* wsb = (char*)d_ws;
  size_t off = 0;
  auto alloc = [&](size_t bytes) -> void* {
    void* p = wsb + off;
    off = (off + bytes + 255) & ~(size_t)255;
    return p;
  };
  int*      etype  = (int*)alloc((size_t)E * sizeof(int));
  float*    hA     = (float*)alloc((size_t)N * 64 * sizeof(float));
  float*    hB     = (float*)alloc((size_t)N * 64 * sizeof(float));
  float*    trbuf  = (float*)alloc((size_t)4 * N * 64 * sizeof(float));
  float*    aggr   = (float*)alloc((size_t)N * 64 * sizeof(float));
  float*    mu     = (float*)alloc(256 * sizeof(float));
  float*    rs     = (float*)alloc(256 * sizeof(float));
  float*    eN     = (float*)alloc((size_t)N * sizeof(float));
  float*    exN    = (float*)alloc((size_t)N * sizeof(float));
  float*    aN     = (float*)alloc((size_t)N * sizeof(float));
  unsigned* emaxU  = (unsigned*)alloc((size_t)B * sizeof(unsigned));
  float*    denom  = (float*)alloc((size_t)B * sizeof(float));
  float*    sstate = (float*)alloc((size_t)4 * B * 64 * sizeof(float));  // h0,c0,h1,c1
  float*    qstar  = (float*)alloc((size_t)B * 128 * sizeof(float));
  float*    rbuf   = (float*)alloc((size_t)B * 64 * sizeof(float));
  float*    x1b    = (float*)alloc((size_t)B * 128 * sizeof(float));
  float*    x2b    = (float*)alloc((size_t)B * 128 * sizeof(float));
  float*    h2A    = (float*)alloc((size_t)N2 * 64 * sizeof(float));
  float*    h2B    = (float*)alloc((size_t)N2 * 64 * sizeof(float));
  float*    hsum   = (float*)alloc((size_t)N2 * 64 * sizeof(float));
  float*    cnt    = (float*)alloc((size_t)N2 * sizeof(float));
  float*    zb     = (float*)alloc((size_t)B * 256 * sizeof(float));
  float*    z0b    = (float*)alloc((size_t)B * 128 * sizeof(float));
  float*    z1b    = (float*)alloc((size_t)B * 64 * sizeof(float));
  float*    h0 = sstate, *c0 = sstate + (size_t)B * 64;
  float*    h1 = sstate + (size_t)2 * B * 64, *c1 = sstate + (size_t)3 * B * 64;

  // ---- MGGC (3 inner GRU-conv iterations), returns the live buffer ----
  auto mggc = [&](float* hin, float* htmp, int layer, int Nn, const int* sP, const int* dP,
                  int Ecnt, const int* etP, const float* ntRowP, const int* ntypeP,
                  int Tuse) -> float* {
    float* cur = hin; float* oth = htmp;
    for (int l = 0; l < 3; ++l) {
      const float* Wl = conv_w + ((size_t)layer * 3 + l) * 4 * 64 * 64;
      dim3 gT((unsigned)(Nn / 16), 4, (unsigned)Tuse);
      k_transform<<<gT, 32, 0, stream>>>(cur, Wl, trbuf, Nn);
      k_zero<<<gdiv((size_t)Nn * 64, TPB), TPB, 0, stream>>>(aggr, (size_t)Nn * 64);
      k_scatter<<<gdiv((size_t)Ecnt * 16, TPB), TPB, 0, stream>>>(
          trbuf, sP, dP, etP, ntRowP, ntypeP, aggr, Ecnt, Nn);
      dim3 gG((unsigned)(Nn / 16), 4);
      k_gru<<<gG, 32, 0, stream>>>(aggr, cur,
                                   conv_wih + (size_t)layer * 192 * 64,
                                   conv_whh + (size_t)layer * 192 * 64,
                                   conv_bih + (size_t)layer * 192,
                                   conv_bhh + (size_t)layer * 192, oth, Nn);
      float* t2 = cur; cur = oth; oth = t2;
    }
    return cur;
  };

  // ---- Set2Set ----
  auto run_s2s = [&](const float* xn, const int* bat, int Nn, int which, float* xout) {
    const float* wi0 = s2s_w[which][0]; const float* wh0 = s2s_w[which][1];
    const float* bb0 = s2s_w[which][2]; const float* wi1 = s2s_w[which][3];
    const float* wh1 = s2s_w[which][4]; const float* bb1 = s2s_w[which][5];
    k_zero<<<gdiv((size_t)B * 128, TPB), TPB, 0, stream>>>(qstar, (size_t)B * 128);
    k_zero<<<gdiv((size_t)4 * B * 64, TPB), TPB, 0, stream>>>(sstate, (size_t)4 * B * 64);
    for (int s = 0; s < 5; ++s) {
      k_lstm<<<B, 256, 0, stream>>>(qstar, 128, h0, c0, wi0, wh0, bb0);
      k_lstm<<<B, 256, 0, stream>>>(h0, 64, h1, c1, wi1, wh1, bb1);
      k_att_dot<<<gdiv((size_t)Nn, TPB), TPB, 0, stream>>>(xn, h1, bat, eN, Nn);
      k_zero_u<<<gdiv((size_t)B, TPB), TPB, 0, stream>>>(emaxU, B);
      k_segmax<<<gdiv((size_t)Nn, TPB), TPB, 0, stream>>>(eN, bat, emaxU, Nn);
      k_zero<<<gdiv((size_t)B, TPB), TPB, 0, stream>>>(denom, (size_t)B);
      k_expden<<<gdiv((size_t)Nn, TPB), TPB, 0, stream>>>(eN, emaxU, bat, exN, denom, Nn);
      k_attw<<<gdiv((size_t)Nn, TPB), TPB, 0, stream>>>(exN, denom, bat, aN, Nn);
      k_zero<<<gdiv((size_t)B * 64, TPB), TPB, 0, stream>>>(rbuf, (size_t)B * 64);
      k_wsum<<<gdiv((size_t)Nn * 64, TPB), TPB, 0, stream>>>(aN, xn, bat, rbuf, Nn);
      k_qstar<<<gdiv((size_t)B * 128, TPB), TPB, 0, stream>>>(h1, rbuf, qstar, B);
    }
    hipMemcpyAsync(xout, qstar, (size_t)B * 128 * sizeof(float),
                   hipMemcpyDeviceToDevice, stream);
  };

  // ===== pipeline =====
  k_etype<<<gdiv((size_t)E, TPB), TPB, 0, stream>>>(edgeattr, etype, E);
  hipMemcpyAsync(hA, x, (size_t)N * 64 * sizeof(float), hipMemcpyDeviceToDevice, stream);

  // level-1: 3 conv layers, each followed by BN + ReLU
  float* hcur = hA;
  for (int i = 0; i < 3; ++i) {
    float* halt = (hcur == hA) ? hB : hA;
    hcur = mggc(hcur, halt, i, N, srcP, dstP, E, etype, conv_nt + (size_t)i * 9, node_type, 4);
    k_colstats<<<64, TPB, 0, stream>>>(hcur, N, 64, mu, rs);
    k_bn_apply<<<gdiv((size_t)N * 64, TPB), TPB, 0, stream>>>(
        hcur, mu, rs, bn_g + (size_t)i * 64, bn_b + (size_t)i * 64, 1, (size_t)N * 64, 64);
  }

  // set2set over level-1
  run_s2s(hcur, batch, N, 0, x1b);

  // avg_pool level-1 -> level-2
  k_zero<<<gdiv((size_t)N2 * 64, TPB), TPB, 0, stream>>>(hsum, (size_t)N2 * 64);
  k_zero<<<gdiv((size_t)N2, TPB), TPB, 0, stream>>>(cnt, (size_t)N2);
  k_apool_scatter<<<gdiv((size_t)A * 64, TPB), TPB, 0, stream>>>(hcur, assign_src, assign_dst,
                                                                 hsum, cnt, A);
  k_apool_div<<<gdiv((size_t)N2 * 64, TPB), TPB, 0, stream>>>(h2A, hsum, cnt, N2);

  // level-2: 2 conv layers (etype=0, coeff=1), ReLU after each
  float* h2cur = h2A;
  for (int i = 3; i < 5; ++i) {
    float* h2alt = (h2cur == h2A) ? h2B : h2A;
    h2cur = mggc(h2cur, h2alt, i, N2, src2, dst2, E2, nullptr, nullptr, nullptr, 1);
    k_relu<<<gdiv((size_t)N2 * 64, TPB), TPB, 0, stream>>>(h2cur, (size_t)N2 * 64);
  }

  // set2set over level-2
  run_s2s(h2cur, batch2, N2, 1, x2b);

  // ===== head =====
  k_concat<<<gdiv((size_t)B * 256, TPB), TPB, 0, stream>>>(x1b, x2b, zb, B);
  k_colstats<<<256, TPB, 0, stream>>>(zb, B, 256, mu, rs);
  k_bn_apply<<<gdiv((size_t)B * 256, TPB), TPB, 0, stream>>>(zb, mu, rs, prebn_g, prebn_b, 0,
                                                             (size_t)B * 256, 256);
  k_fc<<<gdiv((size_t)B * 128, TPB), TPB, 0, stream>>>(zb, fc_w0, fc_b0, z0b, B, 256, 128);
  k_colstats<<<128, TPB, 0, stream>>>(z0b, B, 128, mu, rs);
  k_bn_apply<<<gdiv((size_t)B * 128, TPB), TPB, 0, stream>>>(z0b, mu, rs, fcbn_g0, fcbn_b0, 1,
                                                             (size_t)B * 128, 128);
  k_fc<<<gdiv((size_t)B * 64, TPB), TPB, 0, stream>>>(z0b, fc_w1, fc_b1, z1b, B, 128, 64);
  k_colstats<<<64, TPB, 0, stream>>>(z1b, B, 64, mu, rs);
  k_bn_apply<<<gdiv((size_t)B * 64, TPB), TPB, 0, stream>>>(z1b, mu, rs, fcbn_g1, fcbn_b1, 1,
                                                            (size_t)B * 64, 64);
  k_fc<<<gdiv((size_t)B, TPB), TPB, 0, stream>>>(z1b, fc_w2, fc_b2, (float*)d_out, B, 64, 1);
}